// Router_14860586844369
// MI455X (gfx1250) — compile-verified
//
#include <hip/hip_runtime.h>
#include <hip/hip_bf16.h>

typedef __attribute__((ext_vector_type(2))) float v2f;
typedef __attribute__((ext_vector_type(4))) float v4f;
typedef __attribute__((ext_vector_type(8))) float v8f;

#define ROUTER_D 1024
#define ROUTER_E 64
#define TOK_PER_WG 128       // 4 waves * 32 tokens each
#define SROW 68              // LDS row stride: 16B-aligned rows, conflict-free acc stores

__global__ __launch_bounds__(128) void
Router_14860586844369_kernel(const float* __restrict__ H,   // [T, 1024]
                             const float* __restrict__ W,   // [64, 1024]
                             float* __restrict__ outProbs,  // [T, 2]
                             int*   __restrict__ outIdx,    // [T, 2]
                             float* __restrict__ outLogits, // [T, 64]
                             int T)
{
    __shared__ float sm[TOK_PER_WG * SROW];

    const int tid   = threadIdx.x;
    const int wave  = tid >> 5;
    const int lane  = tid & 31;
    const int half  = lane >> 4;   // 0: lanes 0-15, 1: lanes 16-31
    const int l16   = lane & 15;
    const int wgTok = blockIdx.x * TOK_PER_WG;
    const int rowBase = wgTok + wave * 32;   // wave owns two 16-row M-tiles

    // ---------------- GEMM: logits = H * W^T via V_WMMA_F32_16X16X4_F32 ----
    // A frag (16x4 f32): lane%16 = M row, K = 2*half + vgpr
    // B frag (4x16 f32): lane%16 = N col, K = 2*half + vgpr; B[k][n] = W[n][k]
    v8f acc[2][4] = {};

    const float* a0 = H + (size_t)(rowBase +  0 + l16) * ROUTER_D + 2 * half;
    const float* a1 = H + (size_t)(rowBase + 16 + l16) * ROUTER_D + 2 * half;
    const float* bp[4];
#pragma unroll
    for (int n = 0; n < 4; ++n)
        bp[n] = W + (size_t)(16 * n + l16) * ROUTER_D + 2 * half;

#pragma unroll 4
    for (int kb = 0; kb < ROUTER_D; kb += 4) {
        v2f af0 = __builtin_nontemporal_load((const v2f*)(a0 + kb)); // stream-once
        v2f af1 = __builtin_nontemporal_load((const v2f*)(a1 + kb));
        v2f wf[4];
#pragma unroll
        for (int n = 0; n < 4; ++n)
            wf[n] = *(const v2f*)(bp[n] + kb);   // W is L1/L2-resident
#pragma unroll
        for (int n = 0; n < 4; ++n) {
            acc[0][n] = __builtin_amdgcn_wmma_f32_16x16x4_f32(false, af0, false, wf[n],
                                                              (short)0, acc[0][n], false, false);
            acc[1][n] = __builtin_amdgcn_wmma_f32_16x16x4_f32(false, af1, false, wf[n],
                                                              (short)0, acc[1][n], false, false);
        }
    }

    // C/D layout: VGPR v -> M = v + 8*half, N = lane%16 (+16*tile)
#pragma unroll
    for (int r = 0; r < 2; ++r) {
#pragma unroll
        for (int v = 0; v < 8; ++v) {
            const int m = wave * 32 + r * 16 + v + 8 * half;
            float* row = sm + m * SROW + l16;
            row[ 0] = acc[r][0][v];
            row[16] = acc[r][1][v];
            row[32] = acc[r][2][v];
            row[48] = acc[r][3][v];
        }
    }
    __syncthreads();

    // ---------------- coalesced b128 writeback of router_logits -------------
#pragma unroll
    for (int i = tid; i < TOK_PER_WG * ROUTER_E / 4; i += 128) {
        const int t  = i >> 4;          // token within WG
        const int e4 = (i & 15) * 4;    // expert group of 4
        const float* r = sm + t * SROW + e4;
        v4f vv = { r[0], r[1], r[2], r[3] };
        ((v4f*)outLogits)[(size_t)wgTok * (ROUTER_E / 4) + i] = vv;
    }

    // ---------------- softmax over E=64 + top-2 + renormalize ---------------
    {
        const float* row = sm + tid * SROW;

        // argmax (strict > => lowest index on ties, matches jax top_k)
        float m0 = row[0];
        int   i0 = 0;
#pragma unroll
        for (int e = 1; e < ROUTER_E; ++e) {
            float v = row[e];
            if (v > m0) { m0 = v; i0 = e; }
        }
        // second argmax excluding i0
        float m1 = -3.402823466e38f;
        int   i1 = 0;
#pragma unroll
        for (int e = 0; e < ROUTER_E; ++e) {
            if (e == i0) continue;
            float v = row[e];
            if (v > m1) { m1 = v; i1 = e; }
        }
        // softmax denominator in fp32
        float sum = 0.0f;
#pragma unroll
        for (int e = 0; e < ROUTER_E; ++e)
            sum += __expf(row[e] - m0);

        const float inv = 1.0f / sum;
        const float p0  = inv;                    // exp(m0 - m0) * inv
        const float p1  = __expf(m1 - m0) * inv;
        const float rn  = 1.0f / (p0 + p1 + 1e-8f);

        const size_t tok = (size_t)(wgTok + tid);
        outProbs[tok * 2 + 0] = p0 * rn;
        outProbs[tok * 2 + 1] = p1 * rn;
        outIdx  [tok * 2 + 0] = i0;
        outIdx  [tok * 2 + 1] = i1;
    }
}

extern "C" void kernel_launch(void* const* d_in, const int* in_sizes, int n_in,
                              void* d_out, int out_size, void* d_ws, size_t ws_size,
                              hipStream_t stream) {
    const float* H = (const float*)d_in[0];   // [B,S,D] fp32
    const float* W = (const float*)d_in[1];   // [E,D]   fp32

    const int T = in_sizes[0] / ROUTER_D;     // B*S = 32768

    float* outProbs  = (float*)d_out;                       // [T,2] fp32
    int*   outIdx    = (int*)((float*)d_out + (size_t)T*2); // [T,2] int32
    float* outLogits = (float*)d_out + (size_t)T * 4;       // [T,64] fp32

    const int grid = T / TOK_PER_WG;          // 256 workgroups
    Router_14860586844369_kernel<<<grid, 128, 0, stream>>>(H, W, outProbs, outIdx, outLogits, T);
}